// RoIPool_52329881534703
// MI455X (gfx1250) — compile-verified
//
#include <hip/hip_runtime.h>
#include <cstdint>
#include <math.h>

// RoI max pool for MI455X (gfx1250, wave32).
// features: (B=2, C=256, H=56, W=56) f32 ; rois: (B, N=256, 4) f32
// out: (B, N, C, 7, 7) f32
//
// Per block: one (b, n, 32-channel slab). Row bands of the ROI are staged
// into LDS with CDNA5 async-to-LDS DMA ops (ASYNCcnt path), then 7 waves
// reduce the 7 x-bins for 32 channels each, conflict-free in LDS.
// Staging walks (channel-slice, row, granule) with pure nested loops:
// no runtime-divisor integer division, lanes map to consecutive 8B granules.

#define OHH 7
#define OWW 7
#define BB 2
#define CC 256
#define HH 56
#define WW 56
#define NN 256
#define CBLK 32
// channel stride in dwords: even (8B-aligned b64 targets) and 506 % 64 = 58
// -> 32 lanes with lane==channel map to 32 distinct banks (conflict-free).
#define CH_STRIDE 506

__global__ __launch_bounds__(256) void roipool_gfx1250_kernel(
    const float* __restrict__ feat,
    const float* __restrict__ rois,
    float* __restrict__ out)
{
    __shared__ float lds[CBLK * CH_STRIDE];   // 64768 bytes

    const int cb  = blockIdx.x;     // channel block 0..7
    const int n   = blockIdx.y;     // roi
    const int b   = blockIdx.z;     // batch
    const int tid = threadIdx.x;
    const int c0  = cb * CBLK;

    const float* r = rois + (size_t)(b * NN + n) * 4;
    const float rx1 = r[0], ry1 = r[1], rx2 = r[2], ry2 = r[3];

    int x1 = max((int)floorf(rx1 * (float)WW), 0);
    int y1 = max((int)floorf(ry1 * (float)HH), 0);
    int x2 = min((int)floorf(rx2 * (float)WW), WW - 1);
    int y2 = min((int)floorf(ry2 * (float)HH), HH - 1);
    const bool empty = (y2 < y1) || (x2 < x1);

    float* ob = out + ((size_t)(b * NN + n) * CC + c0) * (OHH * OWW);

    if (empty) {
        // reference zeroes empty ROIs
        for (int i = tid; i < CBLK * OHH * OWW; i += 256) ob[i] = 0.0f;
        return;
    }

    const int ylo = min(y1, HH - 1);
    const int xlo = min(x1, WW - 1);
    const int Ly  = max(y2 - ylo + 1, 1);
    const int Lx  = max(x2 - xlo + 1, 1);

    // 8-byte granules covering [xlo, x2] (reduce only reads inside this range)
    const int gxs = xlo >> 1;
    const int gxe = x2 >> 1;

    const float* fb = feat + (size_t)(b * CC + c0) * (HH * WW);

    const int lane = tid & 31;   // wave32 lane
    const int slot = tid >> 5;   // 0..7 ; slots 0..6 -> ow bins in reduce

    for (int oh = 0; oh < OHH; ++oh) {
        const int ys = ylo + (oh * Ly) / OHH;
        const int ye = ylo + (((oh + 1) * Ly + OHH - 1) / OHH) - 1;
        const int yl = ye - ys + 1;      // <= 9 rows

        // ---- async stage: 32 ch x yl rows x [gxs..gxe] granules (b64) ----
        // slot covers channels {slot, slot+8, slot+16, slot+24};
        // lanes cover consecutive granules -> coalesced 8B x up-to-28 per op.
        for (int cc2 = slot; cc2 < CBLK; cc2 += 8) {
            const float* fc = fb + (size_t)cc2 * (HH * WW);
            const int lbase = cc2 * CH_STRIDE;
            for (int rr = 0; rr < yl; ++rr) {
                const float* frow = fc + (size_t)(ys + rr) * WW;
                const int lrow = lbase + rr * WW;
                for (int g = gxs + lane; g <= gxe; g += 32) {
                    unsigned laddr = (unsigned)(uintptr_t)&lds[lrow + g * 2];
                    asm volatile("global_load_async_to_lds_b64 %0, %1, off"
                                 :: "v"(laddr),
                                    "v"((unsigned long long)(uintptr_t)(frow + g * 2))
                                 : "memory");
                }
            }
        }
        asm volatile("s_wait_asynccnt 0" ::: "memory");
        __syncthreads();

        // ---- reduce: wave `slot` handles ow = slot for all 32 channels ----
        if (slot < OWW) {
            const int ow = slot;
            const int xs = xlo + (ow * Lx) / OWW;
            const int xe = xlo + (((ow + 1) * Lx + OWW - 1) / OWW) - 1;
            float m = -INFINITY;
            const int base = lane * CH_STRIDE;   // lane == channel
            for (int rr = 0; rr < yl; ++rr) {
                const int rb = base + rr * WW;
                #pragma unroll 1
                for (int x = xs; x <= xe; ++x)
                    m = fmaxf(m, lds[rb + x]);
            }
            ob[(size_t)lane * (OHH * OWW) + oh * OWW + ow] = m;
        }
        __syncthreads();   // protect LDS before next band is staged
    }
}

extern "C" void kernel_launch(void* const* d_in, const int* in_sizes, int n_in,
                              void* d_out, int out_size, void* d_ws, size_t ws_size,
                              hipStream_t stream) {
    const float* feat = (const float*)d_in[0];
    const float* rois = (const float*)d_in[1];
    float* out = (float*)d_out;
    dim3 grid(CC / CBLK, NN, BB);   // (8, 256, 2)
    roipool_gfx1250_kernel<<<grid, 256, 0, stream>>>(feat, rois, out);
}